// DirectionalCurvatureLoss_17428977287927
// MI455X (gfx1250) — compile-verified
//
#include <hip/hip_runtime.h>

typedef __attribute__((ext_vector_type(2))) float v2f;
typedef __attribute__((ext_vector_type(8))) float v8f;

#define IMG    1024
#define NBATCH 16
#define TW     128         // tile width  (8 waves x 16-pixel strips)
#define TH     32          // tile height (rows per workgroup)
#define LW     (TW + 2)    // 130, LDS tile pitch (halo)
#define LH     (TH + 2)    // 34
#define CELLF  10.0f
#define EPSF   1e-8f

// ds_swizzle_b32 group-of-32 encoding: offset = xor<<10 | or<<5 | and
// SWAPX16: exchanges lanes L <-> L^16 in one LDS-pipe op, no exec games.
#define SWZ_SWAPX16 0x401F

__device__ __forceinline__ float swap16(float x) {
    return __builtin_bit_cast(
        float, __builtin_amdgcn_ds_swizzle(__builtin_bit_cast(int, x), SWZ_SWAPX16));
}

// 1-ulp hardware transcendentals: denominators are (x + 1e-8), x >= 0, feeding
// a mean over 16.7M pixels -> approximation error ~1e-7 relative, negligible.
// Avoids the ~10-instruction IEEE-precise sqrt/div expansions seen in the asm.
__device__ __forceinline__ float fast_sqrt(float x) { return __builtin_amdgcn_sqrtf(x); }
__device__ __forceinline__ float fast_rcp(float x)  { return __builtin_amdgcn_rcpf(x); }

// 5 stencils x 9 taps (row-major dy,dx in {-1,0,1}), padded to 16x12 with zeros
// so WMMA A-fragment lookups never need a branch (pad taps multiply as 0).
__constant__ float WT[16][12] = {
    // sobel_x / 8  -> p
    {-0.125f, 0.f, 0.125f, -0.25f, 0.f, 0.25f, -0.125f, 0.f, 0.125f, 0.f, 0.f, 0.f},
    // sobel_y / 8  -> q
    {-0.125f, -0.25f, -0.125f, 0.f, 0.f, 0.f, 0.125f, 0.25f, 0.125f, 0.f, 0.f, 0.f},
    // k_xx / 3     -> r
    {1.f/3.f, -2.f/3.f, 1.f/3.f, 1.f/3.f, -2.f/3.f, 1.f/3.f, 1.f/3.f, -2.f/3.f, 1.f/3.f, 0.f, 0.f, 0.f},
    // k_xy / 4     -> s
    {-0.25f, 0.f, 0.25f, 0.f, 0.f, 0.f, 0.25f, 0.f, -0.25f, 0.f, 0.f, 0.f},
    // k_yy / 3     -> t
    {1.f/3.f, 1.f/3.f, 1.f/3.f, -2.f/3.f, -2.f/3.f, -2.f/3.f, 1.f/3.f, 1.f/3.f, 1.f/3.f, 0.f, 0.f, 0.f},
    // channels 5..15 implicitly zero
};

__global__ __launch_bounds__(256)
void curv_loss_kernel(const float* __restrict__ pred,
                      const float* __restrict__ targ,
                      float* __restrict__ ws) {
    __shared__ float ldsP[LH * LW];
    __shared__ float ldsT[LH * LW];
    __shared__ float red[3][8];

    const int tid = threadIdx.x;
    const size_t base = (size_t)blockIdx.z * (size_t)(IMG * IMG);
    const int x0 = blockIdx.x * TW;
    const int y0 = blockIdx.y * TH;

    // ---- cooperative halo-tile load (SAME padding = zeros) ----
    for (int i = tid; i < LH * LW; i += 256) {
        int row = i / LW;
        int col = i - row * LW;
        int gy = y0 - 1 + row;
        int gx = x0 - 1 + col;
        bool in = (gy >= 0) & (gy < IMG) & (gx >= 0) & (gx < IMG);
        size_t gi = base + (size_t)gy * IMG + gx;
        ldsP[i] = in ? pred[gi] : 0.f;
        ldsT[i] = in ? targ[gi] : 0.f;
    }
    __syncthreads();

    const int lane  = tid & 31;
    const int wave  = tid >> 5;
    const int half  = lane >> 4;   // 0: lanes 0-15, 1: lanes 16-31
    const int lid   = lane & 15;   // A: channel index, B/D: pixel index
    const int strip = wave * 16;   // 16-pixel-wide strip per wave

    // ---- A fragments: 16(ch) x 4(tap) constant weights ----
    // layout (ISA 7.12.2): lanes0-15 M=lane; VGPR j holds K = 4c + j + 2*half
    v2f A0 = { WT[lid][0 + 2*half], WT[lid][1 + 2*half] };
    v2f A1 = { WT[lid][4 + 2*half], WT[lid][5 + 2*half] };
    v2f A2 = { WT[lid][8 + 2*half], WT[lid][9 + 2*half] }; // taps >=9 are zero

    // ---- per-lane LDS offsets for B fragments (4x16 im2col: tap x pixel) ----
    auto mkoff = [&](int k) -> int {
        if (k > 8) k = 8;                  // clamped: weight is 0 there anyway
        int dy = k / 3, dx = k - 3 * dy;
        return dy * LW + strip + lid + dx; // tile (0,0) == global (y0-1, x0-1)
    };
    const int b00 = mkoff(0 + 2*half);
    const int b01 = mkoff(1 + 2*half);
    const int b10 = mkoff(4 + 2*half);
    const int b11 = mkoff(5 + 2*half);
    const int b20 = mkoff(8 + 2*half);     // k in {8,10}; only k=8 has weight

    float accP = 0.f, accL = 0.f, accM = 0.f;

    for (int it = 0; it < TH; ++it) {
        const int rb = it * LW;

        // ---- pred conv: D = W(16x12pad) x im2col via 3 chained fp32 WMMAs ----
        v2f B0, B1, B2;
        B0[0] = ldsP[rb + b00]; B0[1] = ldsP[rb + b01];
        B1[0] = ldsP[rb + b10]; B1[1] = ldsP[rb + b11];
        B2[0] = ldsP[rb + b20]; B2[1] = 0.f;
        v8f dP = {};
        dP = __builtin_amdgcn_wmma_f32_16x16x4_f32(false, A0, false, B0, (short)0, dP, false, false);
        dP = __builtin_amdgcn_wmma_f32_16x16x4_f32(false, A1, false, B1, (short)0, dP, false, false);
        dP = __builtin_amdgcn_wmma_f32_16x16x4_f32(false, A2, false, B2, (short)0, dP, false, false);

        // ---- target conv ----
        B0[0] = ldsT[rb + b00]; B0[1] = ldsT[rb + b01];
        B1[0] = ldsT[rb + b10]; B1[1] = ldsT[rb + b11];
        B2[0] = ldsT[rb + b20]; B2[1] = 0.f;
        v8f dT = {};
        dT = __builtin_amdgcn_wmma_f32_16x16x4_f32(false, A0, false, B0, (short)0, dT, false, false);
        dT = __builtin_amdgcn_wmma_f32_16x16x4_f32(false, A1, false, B1, (short)0, dT, false, false);
        dT = __builtin_amdgcn_wmma_f32_16x16x4_f32(false, A2, false, B2, (short)0, dT, false, false);

        // D layout: VGPR v, lanes0-15 = channel v for pixel=lane.
        // Branchless: swizzle target channels into the upper half-wave, then
        // cndmask-select so lanes0-15 = pred, lanes16-31 = target (same pixel).
        const float t0 = swap16(dT[0]);
        const float t1 = swap16(dT[1]);
        const float t2 = swap16(dT[2]);
        const float t3 = swap16(dT[3]);
        const float t4 = swap16(dT[4]);
        const float d0 = half ? t0 : dP[0];
        const float d1 = half ? t1 : dP[1];
        const float d2 = half ? t2 : dP[2];
        const float d3 = half ? t3 : dP[3];
        const float d4 = half ? t4 : dP[4];

        // ---- curvature math (fp32; hw sqrt/rcp, ~1 ulp) ----
        const float p = d0 * (1.f / CELLF);
        const float q = d1 * (1.f / CELLF);
        const float r = d2 * (1.f / (CELLF * CELLF));
        const float s = d3 * (1.f / (CELLF * CELLF));
        const float t = d4 * (1.f / (CELLF * CELLF));
        const float p2 = p * p, q2 = q * q, pq = p * q;
        const float den1 = p2 + q2;
        const float sq1p = fast_sqrt(1.f + den1);
        const float sqd  = fast_sqrt(den1);
        float prof = (r * p2 + 2.f * s * pq + t * q2) * fast_rcp(den1 * sq1p + EPSF);
        float plan = (r * q2 - 2.f * s * pq + t * p2) * fast_rcp(den1 * sqd + EPSF);
        float mcur = ((1.f + q2) * r - 2.f * pq * s + (1.f + p2) * t)
                     * fast_rcp(2.f * (1.f + den1) * sq1p + EPSF);
        if (den1 < EPSF) { prof = 0.f; plan = 0.f; }

        // Cross-half diff: every lane computes |pred-targ| for its pixel (each
        // pixel counted twice, once per half -> fold 0.5 into the final scale).
        accP += fabsf(prof - swap16(prof));
        accL += fabsf(plan - swap16(plan));
        accM += fabsf(mcur - swap16(mcur));
    }

    // ---- wave32 butterfly reduction ----
    #pragma unroll
    for (int off = 16; off > 0; off >>= 1) {
        accP += __shfl_xor(accP, off, 32);
        accL += __shfl_xor(accL, off, 32);
        accM += __shfl_xor(accM, off, 32);
    }

    // ---- block reduction in LDS: 8 waves -> 3 atomics per block ----
    if (lane == 0) {
        red[0][wave] = accP;
        red[1][wave] = accL;
        red[2][wave] = accM;
    }
    __syncthreads();
    if (tid < 3) {
        float s = 0.f;
        #pragma unroll
        for (int w = 0; w < 8; ++w) s += red[tid][w];
        atomicAdd(&ws[tid], s);
    }
}

__global__ void zero_ws_kernel(float* ws) {
    if (threadIdx.x < 3) ws[threadIdx.x] = 0.f;
}

__global__ void finalize_kernel(const float* __restrict__ ws, float* __restrict__ out) {
    if (threadIdx.x == 0) {
        // 0.5 factor: every pixel's |diff| was accumulated by both half-waves
        const float invN = 0.5f / ((float)NBATCH * (float)IMG * (float)IMG);
        out[0] = (0.5f * ws[0] + 0.3f * ws[1] + 0.2f * ws[2]) * invN;
    }
}

extern "C" void kernel_launch(void* const* d_in, const int* in_sizes, int n_in,
                              void* d_out, int out_size, void* d_ws, size_t ws_size,
                              hipStream_t stream) {
    (void)in_sizes; (void)n_in; (void)out_size; (void)ws_size;
    const float* pred = (const float*)d_in[0];
    const float* targ = (const float*)d_in[1];
    float* ws  = (float*)d_ws;
    float* out = (float*)d_out;

    zero_ws_kernel<<<1, 32, 0, stream>>>(ws);
    dim3 grid(IMG / TW, IMG / TH, NBATCH);   // 8 x 32 x 16 workgroups
    curv_loss_kernel<<<grid, 256, 0, stream>>>(pred, targ, ws);
    finalize_kernel<<<1, 32, 0, stream>>>(ws, out);
}